// Octree2Col_12824772345910
// MI455X (gfx1250) — compile-verified
//
#include <hip/hip_runtime.h>

// Octree2Col for MI455X (gfx1250): pure gather/transpose, store-BW bound.
// out[c,k,h] = neigh[h,k] >= 0 ? x[c, neigh[h,k]] : 0
//   x     : (C, H)   fp32
//   neigh : (H, K)   int32
//   out   : (C, K, H) fp32   (~1.04 GB -> streamed with NT stores)
//
// Design: x (38.4 MB) and neigh (16.2 MB) are L2-resident (192 MB L2);
// output is write-once, so it is stored non-temporally (TH=NT) as b128
// to avoid evicting x from L2. Each thread handles 4 consecutive h so a
// wave32 writes a contiguous 512B span per channel c.

typedef float v4f __attribute__((ext_vector_type(4)));

__global__ __launch_bounds__(256) void octree2col_kernel(
    const float* __restrict__ x,      // (C, H)
    const int*   __restrict__ neigh,  // (H, K)
    float*       __restrict__ out,    // (C, K, H)
    int H)
{
    constexpr int C = 64;
    constexpr int K = 27;

    const int k = blockIdx.y;
    const int h = (blockIdx.x * blockDim.x + threadIdx.x) * 4;
    if (h >= H) return;

    const size_t outStride = (size_t)K * (size_t)H;  // floats between channels

    if (h + 4 <= H) {
        // ---- vector path: 4 consecutive h per thread ----
        const int i0 = neigh[(size_t)(h + 0) * K + k];
        const int i1 = neigh[(size_t)(h + 1) * K + k];
        const int i2 = neigh[(size_t)(h + 2) * K + k];
        const int i3 = neigh[(size_t)(h + 3) * K + k];

        // branch-free mask handling: clamp to 0, always load, cndmask-select
        const int s0 = i0 >= 0 ? i0 : 0;
        const int s1 = i1 >= 0 ? i1 : 0;
        const int s2 = i2 >= 0 ? i2 : 0;
        const int s3 = i3 >= 0 ? i3 : 0;
        const bool m0 = i0 >= 0;
        const bool m1 = i1 >= 0;
        const bool m2 = i2 >= 0;
        const bool m3 = i3 >= 0;

        const float* xc = x;                              // walks c*H
        float*       o  = out + (size_t)k * H + h;        // walks c*K*H

        #pragma unroll 8
        for (int c = 0; c < C; ++c) {
            const float t0 = xc[s0];   // L2-resident gathers
            const float t1 = xc[s1];
            const float t2 = xc[s2];
            const float t3 = xc[s3];
            v4f v;
            v.x = m0 ? t0 : 0.0f;
            v.y = m1 ? t1 : 0.0f;
            v.z = m2 ? t2 : 0.0f;
            v.w = m3 ? t3 : 0.0f;
            // streaming 128-bit store, TH=NT: don't pollute L2 (x lives there)
            __builtin_nontemporal_store(v, (v4f*)o);
            xc += H;
            o  += outStride;
        }
    } else {
        // ---- scalar tail (only if H % 4 != 0) ----
        for (int j = 0; j < 4 && (h + j) < H; ++j) {
            const int  i = neigh[(size_t)(h + j) * K + k];
            const int  s = i >= 0 ? i : 0;
            const bool m = i >= 0;
            const float* xc = x;
            float*       o  = out + (size_t)k * H + (h + j);
            for (int c = 0; c < C; ++c) {
                const float t = xc[s];
                __builtin_nontemporal_store(m ? t : 0.0f, o);
                xc += H;
                o  += outStride;
            }
        }
    }
}

extern "C" void kernel_launch(void* const* d_in, const int* in_sizes, int n_in,
                              void* d_out, int out_size, void* d_ws, size_t ws_size,
                              hipStream_t stream) {
    const float* x     = (const float*)d_in[0];  // (1, C, H, 1) fp32 == (C,H)
    const int*   neigh = (const int*)d_in[1];    // (H, K) int32
    float*       out   = (float*)d_out;          // (C, K, H) fp32

    constexpr int C = 64;
    constexpr int K = 27;
    const int H = in_sizes[0] / C;               // 150000 for the reference

    const int threads = 256;                     // 8 wave32s -> fills a WGP
    const int hPerBlock = threads * 4;           // 4 h per thread (b128 stores)
    dim3 grid((H + hPerBlock - 1) / hPerBlock, K);

    octree2col_kernel<<<grid, dim3(threads), 0, stream>>>(x, neigh, out, H);
}